// SGGRU_69647189672188
// MI455X (gfx1250) — compile-verified
//
#include <hip/hip_runtime.h>
#include <math.h>

typedef __attribute__((ext_vector_type(2))) float v2f;
typedef __attribute__((ext_vector_type(4))) float v4f;
typedef __attribute__((ext_vector_type(8))) float v8f;
typedef __attribute__((__vector_size__(4 * sizeof(int)))) int v4i;

#define HN 4096           // H == IN == 4096
#define BK1 64            // phase-1 K-chunk
#define LDK1 (BK1 + 4)    // 16B-aligned rows, conflict-free banks
#define BK2 32            // phase-2 K-chunk (double-buffered fits 64KB LDS)
#define LDK2 (BK2 + 4)

#ifndef __has_builtin
#define __has_builtin(x) 0
#endif

#if __has_builtin(__builtin_amdgcn_global_load_async_to_lds_b128)
#define USE_ASYNC 1
#else
#define USE_ASYNC 0
#endif

// Stage 16 bytes global -> LDS (ASYNCcnt-tracked async copy on CDNA5).
__device__ __forceinline__ void stage16(float* l, const float* g) {
#if USE_ASYNC
  __builtin_amdgcn_global_load_async_to_lds_b128(
      (__attribute__((address_space(1))) v4i*)(g),
      (__attribute__((address_space(3))) v4i*)(l),
      /*imm offset=*/0, /*cpol=*/0);
#else
  *(v4f*)l = *(const v4f*)g;
#endif
}

__device__ __forceinline__ void stage_fence() {
#if USE_ASYNC
#if __has_builtin(__builtin_amdgcn_s_wait_asynccnt)
  __builtin_amdgcn_s_wait_asynccnt(0);
#else
  asm volatile("s_wait_asynccnt 0" ::: "memory");
#endif
#endif
}

// D = A(16x4) * B(4x16) + C, fp32 WMMA (wave32)
__device__ __forceinline__ v8f wmma_f32(v2f a, v2f b, v8f c) {
  return __builtin_amdgcn_wmma_f32_16x16x4_f32(
      /*neg_a=*/false, a, /*neg_b=*/false, b,
      /*c_mod=*/(short)0, c, /*reuse_a=*/false, /*reuse_b=*/false);
}

// ---------------------------------------------------------------------------
// Phase 1: f = tanh(x @ W_mlp^T)   [32,4096]
// One block = one 16-column tile; 2 waves = 2 batch M-tiles.
// Double-buffered async staging: load chunk k+1 while computing chunk k.
// ---------------------------------------------------------------------------
__global__ __launch_bounds__(64) void sggru_filter(const float* __restrict__ x,
                                                   const float* __restrict__ Wm,
                                                   float* __restrict__ fbuf) {
  __shared__ __align__(16) float lw[2][16 * LDK1];
  __shared__ __align__(16) float la[2][32 * LDK1];

  const int t     = threadIdx.x;   // 0..63
  const int lane  = t & 31;
  const int mtile = t >> 5;
  const int j0    = blockIdx.x * 16;
  const int mr    = lane & 15;
  const int hi    = lane >> 4;
  const int srow  = t >> 4;        // staging row 0..3
  const int sc4   = t & 15;        // staging float4 column

  const size_t gw0 = (size_t)(j0 + srow) * HN + sc4 * 4;
  const size_t ga0 = (size_t)srow * HN + sc4 * 4;
  const int    lo  = srow * LDK1 + sc4 * 4;

  auto stage = [&](int buf, int kc) {
    const float* g = Wm + gw0 + kc;
    float*       l = &lw[buf][lo];
#pragma unroll
    for (int it = 0; it < 4; ++it) {   // 16 weight rows
      stage16(l, g);
      g += 4 * (size_t)HN;
      l += 4 * LDK1;
    }
    g = x + ga0 + kc;
    l = &la[buf][lo];
#pragma unroll
    for (int it = 0; it < 8; ++it) {   // 32 activation rows
      stage16(l, g);
      g += 4 * (size_t)HN;
      l += 4 * LDK1;
    }
  };

  v8f acc = {};

  stage(0, 0);
  stage_fence();
  __syncthreads();

  for (int kc = 0; kc < HN; kc += BK1) {
    const int cur = (kc / BK1) & 1;
    if (kc + BK1 < HN) stage(cur ^ 1, kc + BK1);  // prefetch next chunk

    const float* lap = &la[cur][(mtile * 16 + mr) * LDK1 + 2 * hi];
    const float* lwp = &lw[cur][mr * LDK1 + 2 * hi];
#pragma unroll
    for (int kb = 0; kb < BK1; kb += 4) {
      v2f a, b;
      a.x = lap[kb]; a.y = lap[kb + 1];
      b.x = lwp[kb]; b.y = lwp[kb + 1];   // B[k][n] = W[j0+n][k]
      acc = wmma_f32(a, b, acc);
    }
    stage_fence();
    __syncthreads();
  }

  // C/D layout: lanes 0-15 -> M = v, lanes 16-31 -> M = 8+v ; N = lane%16
#pragma unroll
  for (int v = 0; v < 8; ++v) {
    const int bm = mtile * 16 + hi * 8 + v;
    fbuf[(size_t)bm * HN + j0 + mr] = tanhf(acc[v]);
  }
}

// ---------------------------------------------------------------------------
// Phase 2: fully fused gates. 8 GEMMs share A-fragments (f, h0, hg0);
// z/r/n accumulate in registers; double-buffered async staging; epilogue fused.
// ---------------------------------------------------------------------------
__global__ __launch_bounds__(64) void sggru_gates(
    const float* __restrict__ fbuf, const float* __restrict__ h0,
    const float* __restrict__ hg0,
    const float* __restrict__ Wxz, const float* __restrict__ Whz,
    const float* __restrict__ Wgpz, const float* __restrict__ Wxr,
    const float* __restrict__ Whr, const float* __restrict__ Wgpr,
    const float* __restrict__ Wxn, const float* __restrict__ Wgpn,
    float* __restrict__ out) {
  __shared__ __align__(16) float lw[2][8][16 * LDK2];  // 2 x 8 weight tiles
  __shared__ __align__(16) float la[2][3][32 * LDK2];  // 2 x {f,h0,hg0}

  const float* Ws[8] = {Wxz, Whz, Wgpz, Wxr, Whr, Wgpr, Wxn, Wgpn};
  const float* As[3] = {fbuf, h0, hg0};

  const int t     = threadIdx.x;
  const int lane  = t & 31;
  const int mtile = t >> 5;
  const int j0    = blockIdx.x * 16;
  const int mr    = lane & 15;
  const int hi    = lane >> 4;
  const int srow  = t >> 3;        // staging row 0..7
  const int sc4   = t & 7;         // staging float4 column (BK2/4 == 8)

  const size_t gw0 = (size_t)(j0 + srow) * HN + sc4 * 4;
  const size_t ga0 = (size_t)srow * HN + sc4 * 4;
  const int    lo  = srow * LDK2 + sc4 * 4;

  auto stage = [&](int buf, int kc) {
#pragma unroll
    for (int w = 0; w < 8; ++w) {        // 16 rows per tile -> 2 steps of 8
      const float* g = Ws[w] + gw0 + kc;
      float*       l = &lw[buf][w][lo];
#pragma unroll
      for (int it = 0; it < 2; ++it) {
        stage16(l, g);
        g += 8 * (size_t)HN;
        l += 8 * LDK2;
      }
    }
#pragma unroll
    for (int m = 0; m < 3; ++m) {        // 32 rows per tile -> 4 steps of 8
      const float* g = As[m] + ga0 + kc;
      float*       l = &la[buf][m][lo];
#pragma unroll
      for (int it = 0; it < 4; ++it) {
        stage16(l, g);
        g += 8 * (size_t)HN;
        l += 8 * LDK2;
      }
    }
  };

  v8f zacc = {}, racc = {}, nacc = {};

  stage(0, 0);
  stage_fence();
  __syncthreads();

  for (int kc = 0; kc < HN; kc += BK2) {
    const int cur = (kc / BK2) & 1;
    if (kc + BK2 < HN) stage(cur ^ 1, kc + BK2);  // prefetch next chunk

    const float* lap0 = &la[cur][0][(mtile * 16 + mr) * LDK2 + 2 * hi];
    const float* lap1 = &la[cur][1][(mtile * 16 + mr) * LDK2 + 2 * hi];
    const float* lap2 = &la[cur][2][(mtile * 16 + mr) * LDK2 + 2 * hi];
    const float* lwp[8];
#pragma unroll
    for (int w = 0; w < 8; ++w) lwp[w] = &lw[cur][w][mr * LDK2 + 2 * hi];

#pragma unroll
    for (int kb = 0; kb < BK2; kb += 4) {
      v2f af, ah, ag;
      af.x = lap0[kb]; af.y = lap0[kb + 1];
      ah.x = lap1[kb]; ah.y = lap1[kb + 1];
      ag.x = lap2[kb]; ag.y = lap2[kb + 1];
      v2f bw[8];
#pragma unroll
      for (int w = 0; w < 8; ++w) {
        bw[w].x = lwp[w][kb];
        bw[w].y = lwp[w][kb + 1];
      }
      zacc = wmma_f32(af, bw[0], zacc);   // f   @ Wxz^T
      zacc = wmma_f32(ah, bw[1], zacc);   // h0  @ Whz^T
      zacc = wmma_f32(ag, bw[2], zacc);   // hg0 @ Wgpz^T
      racc = wmma_f32(af, bw[3], racc);   // f   @ Wxr^T
      racc = wmma_f32(ah, bw[4], racc);   // h0  @ Whr^T
      racc = wmma_f32(ag, bw[5], racc);   // hg0 @ Wgpr^T
      nacc = wmma_f32(af, bw[6], nacc);   // f   @ Wxn^T
      nacc = wmma_f32(ag, bw[7], nacc);   // hg0 @ Wgpn^T
    }
    stage_fence();
    __syncthreads();
  }

  // fused epilogue: z = sig, r = sig, n = tanh(npre + r*h0), hp = lerp
#pragma unroll
  for (int v = 0; v < 8; ++v) {
    const int bm = mtile * 16 + hi * 8 + v;
    const int j  = j0 + mr;
    const float hv = h0[(size_t)bm * HN + j];
    const float z  = 1.0f / (1.0f + expf(-zacc[v]));
    const float r  = 1.0f / (1.0f + expf(-racc[v]));
    const float n  = tanhf(nacc[v] + r * hv);
    out[(size_t)bm * HN + j] = (1.0f - z) * hv + z * n;
  }
}

// ---------------------------------------------------------------------------
extern "C" void kernel_launch(void* const* d_in, const int* in_sizes, int n_in,
                              void* d_out, int out_size, void* d_ws, size_t ws_size,
                              hipStream_t stream) {
  const float* x    = (const float*)d_in[0];
  const float* h0   = (const float*)d_in[1];
  const float* hg0  = (const float*)d_in[2];
  const float* Wm   = (const float*)d_in[3];
  const float* Wxz  = (const float*)d_in[4];
  const float* Whz  = (const float*)d_in[5];
  const float* Wgpz = (const float*)d_in[6];
  const float* Wxr  = (const float*)d_in[7];
  const float* Whr  = (const float*)d_in[8];
  const float* Wgpr = (const float*)d_in[9];
  const float* Wxn  = (const float*)d_in[10];
  const float* Wgpn = (const float*)d_in[11];

  float* fbuf = (float*)d_ws;        // [32,4096] fp32 = 512 KB scratch
  float* out  = (float*)d_out;

  sggru_filter<<<HN / 16, 64, 0, stream>>>(x, Wm, fbuf);
  sggru_gates<<<HN / 16, 64, 0, stream>>>(fbuf, h0, hg0, Wxz, Whz, Wgpz, Wxr,
                                          Whr, Wgpr, Wxn, Wgpn, out);
}